// TensorGNAN_81020263072227
// MI455X (gfx1250) — compile-verified
//
#include <hip/hip_runtime.h>

#define Nn 2048
#define Ff 32
#define Hh 32
#define Oo 8

typedef __attribute__((ext_vector_type(2))) float v2f;
typedef __attribute__((ext_vector_type(8))) float v8f;

// ---------------------------------------------------------------------------
// Kernel 0: build zero-padded weight matrices so the WMMA B-operand loads are
// unconditional (no EXEC-masked divergent loads inside the K-loops).
//   fW2p: [1024][16], cols 0..7 = fW2[k][c], cols 8..15 = 0
//   rW2p: [32][16],   cols 0..7 = rW2[k][c], cols 8..15 = 0
// ---------------------------------------------------------------------------
__global__ void k_pad(const float* __restrict__ fW2, const float* __restrict__ rW2,
                      float* __restrict__ fW2p, float* __restrict__ rW2p)
{
    int t = blockIdx.x * blockDim.x + threadIdx.x;
    const int n1 = Ff * Hh * 16;          // 16384
    const int n2 = Hh * 16;               // 512
    if (t < n1) {
        int k = t >> 4, c = t & 15;
        fW2p[t] = (c < Oo) ? fW2[k * Oo + c] : 0.0f;
    } else if (t < n1 + n2) {
        int u = t - n1;
        int k = u >> 4, c = u & 15;
        rW2p[u] = (c < Oo) ? rW2[k * Oo + c] : 0.0f;
    }
}

// ---------------------------------------------------------------------------
// Kernel 1: f_sum[n,o] = relu(x[n,f]*fW1[f,h]+fb1[f,h]) contracted with
// fW2[f,h,o], plus sum_f fb2[f,o].  GEMM view: H[2048,1024] @ W2f[1024,8].
// One wave per 16-row tile; A operand (16x4 f32) built in registers in the
// native WMMA lane layout.  Software-pipelined: all 32 loads of one
// f-iteration are issued first (clause-grouped, overlapping), then the 8
// WMMAs run back-to-back with the VALU a-operand builds co-executing.
// ---------------------------------------------------------------------------
__global__ __launch_bounds__(256) void k_fsum(const float* __restrict__ x,
    const float* __restrict__ fW1, const float* __restrict__ fb1,
    const float* __restrict__ fW2p, const float* __restrict__ fb2,
    float* __restrict__ fsum)
{
    int tid   = threadIdx.x;
    int wave  = tid >> 5;
    int lane  = tid & 31;
    int tile  = blockIdx.x * 8 + wave;    // 0..255
    int n0    = tile * 16;
    int row   = n0 + (lane & 15);
    int khalf = lane >> 4;
    int col   = lane & 15;

    v8f c = {};
    #pragma unroll 1
    for (int f = 0; f < Ff; ++f) {
        float xv = x[row * Ff + f];

        // ---- issue all loads for this f up front ----
        v2f w[8], bi[8], b[8];
        #pragma unroll
        for (int s = 0; s < 8; ++s) {
            int kA = f * Hh + s * 4 + khalf * 2;   // this lane's K for .x
            int h0 = kA & 31;                       // even -> 8B aligned
            w[s]  = *(const v2f*)(fW1 + f * Hh + h0);   // global_load_b64
            bi[s] = *(const v2f*)(fb1 + f * Hh + h0);   // global_load_b64
            b[s].x = fW2p[(size_t)kA       * 16 + col];
            b[s].y = fW2p[(size_t)(kA + 1) * 16 + col];
        }
        // ---- compute: 8 WMMAs back-to-back ----
        #pragma unroll
        for (int s = 0; s < 8; ++s) {
            v2f a;
            a.x = fmaxf(fmaf(xv, w[s].x, bi[s].x), 0.0f);
            a.y = fmaxf(fmaf(xv, w[s].y, bi[s].y), 0.0f);
            c = __builtin_amdgcn_wmma_f32_16x16x4_f32(false, a, false, b[s],
                                                      (short)0, c, false, false);
        }
    }
    if (col < Oo) {
        float bsum = 0.0f;
        #pragma unroll
        for (int f = 0; f < Ff; ++f) bsum += fb2[f * Oo + col];
        #pragma unroll
        for (int v = 0; v < 8; ++v) {
            int r2 = n0 + v + khalf * 8;  // C layout: VGPR v -> M=v (+8 hi half)
            fsum[(size_t)r2 * Oo + col] = c[v] + bsum;
        }
    }
}

// ---------------------------------------------------------------------------
// Kernel 2 (hotspot, VALU-bound): for each column j accumulate over an
// i-chunk:   A[j,h] += relu(inv*w1[h]+b1[h]) * invn,   r[j] += invn
// with inv = 1/(1+dist[i,j]), invn = (norm==0 ? 1 : 1/norm).
// j striped across lanes -> coalesced row reads.  The h-loop is written as 16
// explicit float2 pairs so it lowers to v_pk_fma_f32 (packed dual-fp32 FMA,
// 2x rate) + v_dual_max_num_f32, halving the VALU op count of the hot loop.
// w1/b1 are uniform-address reads -> scalarized to SGPRs, LICM-hoisted.
// ---------------------------------------------------------------------------
__global__ __launch_bounds__(256) void k_rho(const float* __restrict__ dist,
    const float* __restrict__ norm, const float* __restrict__ rw1,
    const float* __restrict__ rb1, float* __restrict__ A_part,
    float* __restrict__ r_part, int ilen)
{
    int j     = blockIdx.x * blockDim.x + threadIdx.x;   // 0..2047
    int chunk = blockIdx.y;
    int i0    = chunk * ilen;
    int i1    = min(Nn, i0 + ilen);

    // thread-uniform weights -> SGPRs
    v2f w1v[16], b1v[16];
    #pragma unroll
    for (int h2 = 0; h2 < 16; ++h2) {
        w1v[h2].x = rw1[2 * h2];
        w1v[h2].y = rw1[2 * h2 + 1];
        b1v[h2].x = rb1[2 * h2];
        b1v[h2].y = rb1[2 * h2 + 1];
    }

    v2f acc[16];
    #pragma unroll
    for (int h2 = 0; h2 < 16; ++h2) { acc[h2].x = 0.0f; acc[h2].y = 0.0f; }
    float accR = 0.0f;

    #pragma unroll 2
    for (int i = i0; i < i1; ++i) {
        size_t idx = (size_t)i * Nn + j;
        float d    = dist[idx];
        float nm   = norm[idx];
        float inv  = 1.0f / (1.0f + d);
        float invn = (nm == 0.0f) ? 1.0f : (1.0f / nm);
        accR += invn;
        #pragma unroll
        for (int h2 = 0; h2 < 16; ++h2) {
            v2f t;
            t.x = fmaf(inv, w1v[h2].x, b1v[h2].x);     // -> v_pk_fma_f32
            t.y = fmaf(inv, w1v[h2].y, b1v[h2].y);
            t.x = fmaxf(t.x, 0.0f);                    // -> v_dual_max_num_f32
            t.y = fmaxf(t.y, 0.0f);
            acc[h2].x = fmaf(t.x, invn, acc[h2].x);    // -> v_pk_fma_f32
            acc[h2].y = fmaf(t.y, invn, acc[h2].y);
        }
    }
    float* Ap = A_part + ((size_t)chunk * Nn + j) * Hh;
    #pragma unroll
    for (int h2 = 0; h2 < 16; ++h2) {                  // merges to b128 stores
        Ap[2 * h2]     = acc[h2].x;
        Ap[2 * h2 + 1] = acc[h2].y;
    }
    r_part[(size_t)chunk * Nn + j] = accR;
}

// ---------------------------------------------------------------------------
// Kernel 3: one wave per 16-j tile.  Chunk-reduce A, then S = A @ W2rho via
// 8x v_wmma_f32_16x16x4_f32 (B from padded rW2p, unconditional loads), add
// b2rho[c]*r[j], weight by f_sum[j,c], emit per-tile partial (fixed order).
// ---------------------------------------------------------------------------
__global__ __launch_bounds__(32) void k_combine(const float* __restrict__ A_part,
    const float* __restrict__ r_part, const float* __restrict__ fsum,
    const float* __restrict__ rW2p, const float* __restrict__ rb2,
    float* __restrict__ tile_out, int nchunks)
{
    __shared__ float rlds[16];
    __shared__ float plds[16];
    int lane  = threadIdx.x;
    int tile  = blockIdx.x;                 // 0..127
    int jrow  = tile * 16 + (lane & 15);
    int khalf = lane >> 4;
    int col   = lane & 15;

    // chunk-reduce the 16 A-elements this lane feeds into the 8 WMMA steps
    float av[16];
    #pragma unroll
    for (int s = 0; s < 8; ++s) {
        int cbase = s * 4 + khalf * 2;
        float a0 = 0.0f, a1 = 0.0f;
        for (int ch = 0; ch < nchunks; ++ch) {
            const float* p = A_part + ((size_t)ch * Nn + jrow) * Hh;
            a0 += p[cbase];
            a1 += p[cbase + 1];
        }
        av[2 * s]     = a0;
        av[2 * s + 1] = a1;
    }
    if (lane < 16) {
        float rs = 0.0f;
        for (int ch = 0; ch < nchunks; ++ch)
            rs += r_part[(size_t)ch * Nn + tile * 16 + lane];
        rlds[lane] = rs;
    }
    __syncthreads();

    v8f c = {};
    #pragma unroll
    for (int s = 0; s < 8; ++s) {
        int k = s * 4 + khalf * 2;
        v2f a, b;
        a.x = av[2 * s];
        a.y = av[2 * s + 1];
        b.x = rW2p[(size_t)k       * 16 + col];
        b.y = rW2p[(size_t)(k + 1) * 16 + col];
        c = __builtin_amdgcn_wmma_f32_16x16x4_f32(false, a, false, b,
                                                  (short)0, c, false, false);
    }

    if (col < Oo) {
        float b2 = rb2[col];
        float p  = 0.0f;
        #pragma unroll
        for (int v = 0; v < 8; ++v) {
            int jl = v + khalf * 8;                    // local j in tile
            int jj = tile * 16 + jl;
            float s = c[v] + b2 * rlds[jl];            // S[j,col]
            p = fmaf(s, fsum[(size_t)jj * Oo + col], p);
        }
        plds[khalf * 8 + col] = p;
    }
    __syncthreads();
    if (lane < Oo) tile_out[tile * Oo + lane] = plds[lane] + plds[8 + lane];
}

// ---------------------------------------------------------------------------
// Kernel 4: 128x8 -> 8 final reduction (fixed order), writes d_out.
// ---------------------------------------------------------------------------
__global__ void k_final(const float* __restrict__ tile_out, float* __restrict__ out)
{
    int c = threadIdx.x;
    if (c < Oo) {
        float p = 0.0f;
        for (int t = 0; t < 128; ++t) p += tile_out[t * Oo + c];
        out[c] = p;
    }
}

extern "C" void kernel_launch(void* const* d_in, const int* in_sizes, int n_in,
                              void* d_out, int out_size, void* d_ws, size_t ws_size,
                              hipStream_t stream)
{
    const float* x    = (const float*)d_in[0];
    const float* dist = (const float*)d_in[1];
    const float* norm = (const float*)d_in[2];
    const float* fW1  = (const float*)d_in[3];
    const float* fb1  = (const float*)d_in[4];
    const float* fW2  = (const float*)d_in[5];
    const float* fb2  = (const float*)d_in[6];
    const float* rw1  = (const float*)d_in[7];
    const float* rb1  = (const float*)d_in[8];
    const float* rW2  = (const float*)d_in[9];
    const float* rb2  = (const float*)d_in[10];
    float* out = (float*)d_out;

    // workspace layout (floats): fsum | tile_out | fW2p | rW2p | r_part | A_part
    float* ws = (float*)d_ws;
    size_t avail = ws_size / sizeof(float);
    const size_t fsum_n    = (size_t)Nn * Oo;        // 16384
    const size_t tout_n    = (size_t)128 * Oo;       // 1024
    const size_t fW2p_n    = (size_t)Ff * Hh * 16;   // 16384
    const size_t rW2p_n    = (size_t)Hh * 16;        // 512
    const size_t fixed_n   = fsum_n + tout_n + fW2p_n + rW2p_n;
    const size_t per_chunk = (size_t)Nn * Hh + Nn;   // 67584
    long nc = 1;
    if (avail > fixed_n + per_chunk) {
        nc = (long)((avail - fixed_n) / per_chunk);
        if (nc > 16) nc = 16;
        if (nc < 1)  nc = 1;
    }
    int NC   = (int)nc;                // deterministic given ws_size
    int ilen = (Nn + NC - 1) / NC;

    float* fsum   = ws;
    float* t_out  = fsum + fsum_n;
    float* fW2p   = t_out + tout_n;
    float* rW2p   = fW2p + fW2p_n;
    float* r_part = rW2p + rW2p_n;
    float* A_part = r_part + (size_t)NC * Nn;

    k_pad    <<<dim3(66),     dim3(256), 0, stream>>>(fW2, rW2, fW2p, rW2p);
    k_fsum   <<<dim3(32),     dim3(256), 0, stream>>>(x, fW1, fb1, fW2p, fb2, fsum);
    k_rho    <<<dim3(8, NC),  dim3(256), 0, stream>>>(dist, norm, rw1, rb1,
                                                      A_part, r_part, ilen);
    k_combine<<<dim3(128),    dim3(32),  0, stream>>>(A_part, r_part, fsum,
                                                      rW2p, rb2, t_out, NC);
    k_final  <<<dim3(1),      dim3(32),  0, stream>>>(t_out, out);

    (void)in_sizes; (void)n_in; (void)out_size;
}